// FusionModule_38714835206596
// MI455X (gfx1250) — compile-verified
//
#include <hip/hip_runtime.h>

typedef float v2f __attribute__((ext_vector_type(2)));
typedef float v8f __attribute__((ext_vector_type(8)));
typedef unsigned int u32x4 __attribute__((ext_vector_type(4)));
typedef int i32x8 __attribute__((ext_vector_type(8)));
typedef int i32x4 __attribute__((ext_vector_type(4)));

constexpr int NB   = 16;      // batches
constexpr int NC   = 256;     // channels (GEMM K)
constexpr int NPOS = 64 * 64; // spatial positions (GEMM M) = 4096
constexpr int NTAP = 16 * 16; // filter taps (GEMM N) = 256
constexpr int OH   = 49;
constexpr int OW   = 49;
constexpr int MMN  = NB * OH * OW; // 38416
constexpr int KCHUNK = 64;    // K rows of B staged in LDS per pass (64KB)

#if defined(__has_builtin)
#if __has_builtin(__builtin_amdgcn_tensor_load_to_lds) && \
    __has_builtin(__builtin_amdgcn_s_wait_tensorcnt)
#define USE_TDM 1
#endif
#endif

// ---------------------------------------------------------------- zero init
__global__ __launch_bounds__(256) void zero_ws_kernel(float* ws, int n) {
    int i = blockIdx.x * 256 + threadIdx.x;
    if (i < n) ws[i] = 0.0f;
}

// ------------------------------------------------- shift-conv GEMM via WMMA
// Per batch b: H[p, t] = sum_c S[b,c,p] * D[b,c,t]  (M=4096, N=256, K=256)
// then scatter H[(y+i,x+j),(i,j)] -> mm[b,y,x] with f32 atomics.
__global__ __launch_bounds__(256) void conv_wmma_kernel(
    const float* __restrict__ S,   // [NB, NC, 4096]
    const float* __restrict__ D,   // [NB, NC, 256]
    float* __restrict__ mm)        // [NB, 49, 49]
{
    __shared__ __align__(16) float bsh[KCHUNK * NTAP]; // 64 KB chunk of B

    const int b     = blockIdx.x;      // batch
    const int mtile = blockIdx.y;      // 32 tiles of 128 positions
    const int tid   = threadIdx.x;
    const int lane  = tid & 31;
    const int wave  = tid >> 5;        // 8 waves, 16 M-rows each
    const int m0    = mtile * 128 + wave * 16;
    const int prow  = m0 + (lane & 15);            // A-row this lane loads
    const int koffA = (lane < 16) ? 0 : 2;         // A lane-half K offset
    const int koffB0 = (lane < 16) ? 0 : 2;        // B VGPR0 K row
    const int koffB1 = (lane < 16) ? 1 : 3;        // B VGPR1 K row
    const int nlo   = lane & 31 & 15;

    const float* Sb = S + (size_t)b * NC * NPOS;
    const float* Db = D + (size_t)b * NC * NTAP;

    v8f acc[16];
    const v8f vzero = {0.f, 0.f, 0.f, 0.f, 0.f, 0.f, 0.f, 0.f};
#pragma unroll
    for (int i = 0; i < 16; ++i) acc[i] = vzero;

    for (int kc = 0; kc < NC; kc += KCHUNK) {
#if defined(USE_TDM)
        // Tensor Data Mover: DMA a 2D tile (256 x 64 f32) of B into LDS.
        {
            const unsigned int ldsOff =
                (unsigned int)(size_t)(&bsh[0]);        // low 32 bits = LDS offset
            const unsigned long long ga =
                (unsigned long long)(size_t)(Db + (size_t)kc * NTAP);
            u32x4 g0;
            g0.x = 1u;                                   // count=1, user D#
            g0.y = ldsOff;                               // lds_addr [63:32]
            g0.z = (unsigned int)(ga & 0xFFFFFFFFull);   // global_addr lo
            g0.w = (unsigned int)((ga >> 32) & 0x01FFFFFFull) | (2u << 30); // type=2
            i32x8 g1;
            g1[0] = 0x00020000;            // data_size=2 (4B), no multicast/flags
            g1[1] = (int)(256u << 16);     // tensor_dim0 = 256 (bits 79:48)
            g1[2] = (int)(64u  << 16);     // tensor_dim1 = 64  (bits 111:80)
            g1[3] = (int)(256u << 16);     // tile_dim0   = 256 (bits 127:112)
            g1[4] = 64;                    // tile_dim1   = 64  (bits 143:128)
            g1[5] = 256;                   // tensor_dim0_stride = 256 (bits 207:160)
            g1[6] = (int)(16384u << 16);   // tensor_dim1_stride (bits 255:208)
            g1[7] = 0;
            const i32x4 gz4 = {0, 0, 0, 0};              // groups 2/3 unused
            const i32x8 gz8 = {0, 0, 0, 0, 0, 0, 0, 0};
            __builtin_amdgcn_tensor_load_to_lds(g0, g1, gz4, gz4, gz8, 0);
            __builtin_amdgcn_s_wait_tensorcnt(0);
        }
#else
        // fallback: stage 64 K-rows of B (64 x 256 f32) into LDS, 128-bit moves
        {
            const float4* gsrc = (const float4*)(Db + (size_t)kc * NTAP);
#pragma unroll
            for (int t = 0; t < (KCHUNK * NTAP / 4) / 256; ++t) {
                int idx = tid + t * 256;
                ((float4*)bsh)[idx] = gsrc[idx];
            }
        }
#endif
        __syncthreads();

        for (int k4 = 0; k4 < KCHUNK; k4 += 4) {
            // prefetch the A rows two K-steps ahead (16KB-strided stream)
            __builtin_prefetch(Sb + (size_t)(kc + k4 + 8 + koffA) * NPOS + prow, 0, 1);
            // A fragment (16x4 f32): lane<16 holds K=k4+0/1, lane>=16 K=k4+2/3
            v2f a;
            a.x = Sb[(size_t)(kc + k4 + koffA)     * NPOS + prow];
            a.y = Sb[(size_t)(kc + k4 + koffA + 1) * NPOS + prow];
#pragma unroll
            for (int nt = 0; nt < 16; ++nt) {
                v2f bf;
                bf.x = bsh[(k4 + koffB0) * NTAP + nt * 16 + nlo];
                bf.y = bsh[(k4 + koffB1) * NTAP + nt * 16 + nlo];
                acc[nt] = __builtin_amdgcn_wmma_f32_16x16x4_f32(
                    false, a, false, bf, (short)0, acc[nt], false, false);
            }
        }
        __syncthreads();
    }

    // scatter: H[p=(y',x'), t=(i,j)] contributes to mm[y'-i, x'-j]
    float* mmb = mm + b * OH * OW;
    const int mbase = m0 + ((lane < 16) ? 0 : 8);
#pragma unroll
    for (int nt = 0; nt < 16; ++nt) {
        const int t = nt * 16 + nlo;
        const int fi = t >> 4;
        const int fj = t & 15;
#pragma unroll
        for (int r = 0; r < 8; ++r) {
            const int p = mbase + r;
            const int y = (p >> 6) - fi;
            const int x = (p & 63) - fj;
            if ((unsigned)y < (unsigned)OH && (unsigned)x < (unsigned)OW)
                atomicAdd(&mmb[y * OW + x], acc[nt][r]);
        }
    }
}

// ------------------------------------------------------ global mean / invstd
__global__ __launch_bounds__(256) void stats_kernel(const float* __restrict__ mm,
                                                    float* __restrict__ st) {
    __shared__ float ssum[256];
    __shared__ float ssq[256];
    float s = 0.f, q = 0.f;
    for (int i = threadIdx.x; i < MMN; i += 256) {
        float v = mm[i];
        s += v;
        q += v * v;
    }
    ssum[threadIdx.x] = s;
    ssq[threadIdx.x]  = q;
    __syncthreads();
    for (int off = 128; off > 0; off >>= 1) {
        if (threadIdx.x < off) {
            ssum[threadIdx.x] += ssum[threadIdx.x + off];
            ssq[threadIdx.x]  += ssq[threadIdx.x + off];
        }
        __syncthreads();
    }
    if (threadIdx.x == 0) {
        float mean = ssum[0] / (float)MMN;
        float var  = ssq[0] / (float)MMN - mean * mean;
        st[0] = mean;
        st[1] = rsqrtf(var + 1e-5f);
    }
}

// ---------------------------------- normalize + bilinear upsample 49 -> 256
__global__ __launch_bounds__(256) void upsample_kernel(
    const float* __restrict__ mm, const float* __restrict__ st,
    const float* __restrict__ gamma, const float* __restrict__ beta,
    float* __restrict__ out)
{
    int idx = blockIdx.x * 256 + threadIdx.x;
    if (idx >= NB * 256 * 256) return;
    int ox = idx & 255;
    int oy = (idx >> 8) & 255;
    int b  = idx >> 16;

    const float scale = 49.0f / 256.0f;
    float cy = fminf(fmaxf((oy + 0.5f) * scale - 0.5f, 0.f), 48.f);
    float cx = fminf(fmaxf((ox + 0.5f) * scale - 0.5f, 0.f), 48.f);
    int y0 = (int)floorf(cy); int y1 = min(y0 + 1, 48); float wy = cy - (float)y0;
    int x0 = (int)floorf(cx); int x1 = min(x0 + 1, 48); float wx = cx - (float)x0;

    const float* m = mm + b * OH * OW;
    float top = m[y0 * OW + x0] * (1.f - wx) + m[y0 * OW + x1] * wx;
    float bot = m[y1 * OW + x0] * (1.f - wx) + m[y1 * OW + x1] * wx;
    float v = top * (1.f - wy) + bot * wy;
    // normalization is affine -> commutes with bilinear interpolation
    out[idx] = (v - st[0]) * st[1] * gamma[0] + beta[0];
}

extern "C" void kernel_launch(void* const* d_in, const int* in_sizes, int n_in,
                              void* d_out, int out_size, void* d_ws, size_t ws_size,
                              hipStream_t stream) {
    const float* S     = (const float*)d_in[0];
    const float* D     = (const float*)d_in[1];
    const float* gamma = (const float*)d_in[2];
    const float* beta  = (const float*)d_in[3];
    float* mm  = (float*)d_ws;          // 38416 f32 accumulators
    float* st  = mm + MMN;              // mean, invstd
    float* out = (float*)d_out;

    zero_ws_kernel<<<(MMN + 2 + 255) / 256, 256, 0, stream>>>(mm, MMN + 2);
    conv_wmma_kernel<<<dim3(NB, NPOS / 128), 256, 0, stream>>>(S, D, mm);
    stats_kernel<<<1, 256, 0, stream>>>(mm, st);
    upsample_kernel<<<(NB * 256 * 256) / 256, 256, 0, stream>>>(mm, st, gamma, beta, out);
}